// RNN_47751446397285
// MI455X (gfx1250) — compile-verified
//
#include <hip/hip_runtime.h>
#include <hip/hip_bf16.h>
#include <math.h>

// ---------------------------------------------------------------------------
// Persistent fused RNN (encoder scan + decoder) for MI455X / gfx1250.
//  grid = 16 workgroups x 256 threads (8 waves). Each WG owns a 32-column
//  slice of H (and a 16-column slice of O). W_hh^T / W_out^T slices live in
//  LDS for the entire run; h ping-pongs through L2 in d_ws; steps are
//  separated by a device-scope atomic grid barrier.
// ---------------------------------------------------------------------------

typedef float v2f __attribute__((ext_vector_type(2)));
typedef float v8f __attribute__((ext_vector_type(8)));

#define Bv      128
#define Tv      1024
#define Iv      256
#define Hv      512
#define Ov      256
#define OUTL    64
#define NWG     16

#define WMMA_F32(a, b, c) \
  __builtin_amdgcn_wmma_f32_16x16x4_f32(false, (a), false, (b), (short)0, (c), false, false)

__global__ void rnn_sync_init(unsigned* sync) {
  sync[0] = 0u;  // arrive counter
  sync[1] = 0u;  // generation
}

__device__ __forceinline__ void grid_sync(unsigned* cnt, unsigned* gen) {
  __syncthreads();
  if (threadIdx.x == 0) {
    unsigned g = __hip_atomic_load(gen, __ATOMIC_RELAXED, __HIP_MEMORY_SCOPE_AGENT);
    unsigned a = __hip_atomic_fetch_add(cnt, 1u, __ATOMIC_ACQ_REL, __HIP_MEMORY_SCOPE_AGENT);
    if (a == (unsigned)(NWG - 1)) {
      __hip_atomic_store(cnt, 0u, __ATOMIC_RELAXED, __HIP_MEMORY_SCOPE_AGENT);
      __hip_atomic_store(gen, g + 1u, __ATOMIC_RELEASE, __HIP_MEMORY_SCOPE_AGENT);
    } else {
      while (__hip_atomic_load(gen, __ATOMIC_ACQUIRE, __HIP_MEMORY_SCOPE_AGENT) == g) {
        __builtin_amdgcn_s_sleep(2);
      }
    }
  }
  __syncthreads();
}

__global__ __launch_bounds__(256)
void rnn_persistent(const float* __restrict__ x,      // [B,T,I]
                    const int*   __restrict__ lengths,// [B]
                    const float* __restrict__ W_ih,   // [H,I]
                    const float* __restrict__ b_ih,   // [H]
                    const float* __restrict__ W_hh,   // [H,H]
                    const float* __restrict__ b_hh,   // [H]
                    const float* __restrict__ W_out,  // [O,H]
                    const float* __restrict__ b_out,  // [O]
                    float*       __restrict__ out,    // [B,OUTL,O]
                    float*       __restrict__ hbuf0,  // [B,H]
                    float*       __restrict__ hbuf1,  // [B,H]
                    unsigned*    __restrict__ sync) {
  extern __shared__ float smem[];
  float* Wh = smem;             // [32][Hv]  W_hh^T slice, n-major (k contiguous)
  float* Wo = smem + 32 * Hv;   // [16][Hv]  W_out^T slice, n-major

  const int wg   = blockIdx.x;        // 0..15
  const int tid  = threadIdx.x;
  const int lane = tid & 31;
  const int wid  = tid >> 5;          // wave id 0..7 -> M band
  const int half = lane >> 4;         // 0: K+{0,1} / rows M..M+7 ; 1: K+{2,3} / rows M+8..
  const int l15  = lane & 15;

  // ---- Preload weight slices into LDS (resident for the whole run) -------
  {
    const float4* src = (const float4*)(W_hh + (size_t)wg * 32 * Hv);
    float4* dst = (float4*)Wh;
    for (int i = tid; i < (32 * Hv) / 4; i += 256) dst[i] = src[i];
    const float4* srcO = (const float4*)(W_out + (size_t)wg * 16 * Hv);
    float4* dstO = (float4*)Wo;
    for (int i = tid; i < (16 * Hv) / 4; i += 256) dstO[i] = srcO[i];
  }
  // ---- Zero our column slice of h0 ----------------------------------------
  for (int i = tid; i < Bv * 32; i += 256) {
    int r = i >> 5, c = (i & 31) + wg * 32;
    hbuf0[r * Hv + c] = 0.0f;
  }
  __syncthreads();
  grid_sync(&sync[0], &sync[1]);

  // ---- Per-wave / per-lane constants --------------------------------------
  const int   mrow  = wid * 16;            // M band base (batch rows)
  const int   arow  = mrow + l15;          // A-fragment row for this lane
  const int   col0  = wg * 32 + l15;       // subtile 0 column (in H)
  const int   col1  = col0 + 16;           // subtile 1 column
  const float bsum0 = b_ih[col0] + b_hh[col0];
  const float bsum1 = b_ih[col1] + b_hh[col1];

  int lenr[8];
#pragma unroll
  for (int r = 0; r < 8; ++r) lenr[r] = lengths[mrow + 8 * half + r];

  v8f hreg0 = {0.f,0.f,0.f,0.f,0.f,0.f,0.f,0.f};
  v8f hreg1 = hreg0;

  const float* hc = hbuf0;   // current h
  float*       hn = hbuf1;   // next h

  const float* xrow = x + (size_t)arow * Tv * Iv;
  const float* wi0  = W_ih + (size_t)(wg * 32 + l15) * Iv;
  const float* wi1  = W_ih + (size_t)(wg * 32 + 16 + l15) * Iv;
  const float* wh0  = Wh + l15 * Hv;
  const float* wh1  = Wh + (l15 + 16) * Hv;
  const int    kofs = 2 * half;

  // ======================= ENCODER: 1024 sequential steps ==================
  for (int t = 0; t < Tv; ++t) {
    v8f c0 = {0.f,0.f,0.f,0.f,0.f,0.f,0.f,0.f};
    v8f c1 = c0;

    // h @ W_hh^T   (K = 512, fp32 WMMA, B-fragments from LDS)
    const float* ha = hc + (size_t)arow * Hv;
#pragma unroll 8
    for (int kk = 0; kk < Hv; kk += 4) {
      const int kc = kk + kofs;
      v2f a  = *(const v2f*)(ha  + kc);
      v2f b0 = *(const v2f*)(wh0 + kc);
      v2f b1 = *(const v2f*)(wh1 + kc);
      c0 = WMMA_F32(a, b0, c0);
      c1 = WMMA_F32(a, b1, c1);
    }
    // fused x_proj:  x_t @ W_ih^T   (K = 256)
    const float* xa = xrow + (size_t)t * Iv;
#pragma unroll 8
    for (int kk = 0; kk < Iv; kk += 4) {
      const int kc = kk + kofs;
      v2f a  = *(const v2f*)(xa  + kc);
      v2f b0 = *(const v2f*)(wi0 + kc);
      v2f b1 = *(const v2f*)(wi1 + kc);
      c0 = WMMA_F32(a, b0, c0);
      c1 = WMMA_F32(a, b1, c1);
    }
    if (t + 1 < Tv) __builtin_prefetch(xa + Iv, 0, 0);  // next timestep of x

    // epilogue: bias + tanh + length-mask blend, keep h in regs, publish to L2
#pragma unroll
    for (int r = 0; r < 8; ++r) {
      const bool mk = (t < lenr[r]);
      const float v0 = tanhf(c0[r] + bsum0);
      const float v1 = tanhf(c1[r] + bsum1);
      const float n0 = mk ? v0 : hreg0[r];
      const float n1 = mk ? v1 : hreg1[r];
      hreg0[r] = n0;
      hreg1[r] = n1;
      const int row = mrow + 8 * half + r;
      hn[row * Hv + col0] = n0;
      hn[row * Hv + col1] = n1;
    }
    grid_sync(&sync[0], &sync[1]);
    const float* tmp = hc; hc = hn; hn = (float*)tmp;  // ping-pong
  }

  // ======================= DECODER: 64 sequential steps ====================
  const int   ocol = wg * 16 + l15;
  const float bo   = b_out[ocol];
  const float* wo0 = Wo + l15 * Hv;

  for (int s = 0; s < OUTL; ++s) {
    v8f c0 = {0.f,0.f,0.f,0.f,0.f,0.f,0.f,0.f};
    v8f c1 = c0;

    const float* ha = hc + (size_t)arow * Hv;
#pragma unroll 8
    for (int kk = 0; kk < Hv; kk += 4) {
      const int kc = kk + kofs;
      v2f a  = *(const v2f*)(ha  + kc);
      v2f b0 = *(const v2f*)(wh0 + kc);
      v2f b1 = *(const v2f*)(wh1 + kc);
      c0 = WMMA_F32(a, b0, c0);
      c1 = WMMA_F32(a, b1, c1);
    }
#pragma unroll
    for (int r = 0; r < 8; ++r) {
      const float v0 = tanhf(c0[r] + bsum0);
      const float v1 = tanhf(c1[r] + bsum1);
      hreg0[r] = v0;
      hreg1[r] = v1;
      const int row = mrow + 8 * half + r;
      hn[row * Hv + col0] = v0;
      hn[row * Hv + col1] = v1;
    }
    grid_sync(&sync[0], &sync[1]);   // new h now globally visible in hn

    // out_s = h_new @ W_out^T + b_out   (K = 512, B-fragments from LDS)
    v8f co = {0.f,0.f,0.f,0.f,0.f,0.f,0.f,0.f};
    const float* ha2 = hn + (size_t)arow * Hv;
#pragma unroll 8
    for (int kk = 0; kk < Hv; kk += 4) {
      const int kc = kk + kofs;
      v2f a = *(const v2f*)(ha2 + kc);
      v2f b = *(const v2f*)(wo0 + kc);
      co = WMMA_F32(a, b, co);
    }
#pragma unroll
    for (int r = 0; r < 8; ++r) {
      const int row = mrow + 8 * half + r;
      out[(size_t)row * (OUTL * Ov) + s * Ov + ocol] = co[r] + bo;
    }
    const float* tmp = hc; hc = hn; hn = (float*)tmp;  // ping-pong
  }
}

// ---------------------------------------------------------------------------
extern "C" void kernel_launch(void* const* d_in, const int* in_sizes, int n_in,
                              void* d_out, int out_size, void* d_ws, size_t ws_size,
                              hipStream_t stream) {
  const float* x       = (const float*)d_in[0];
  const int*   lengths = (const int*)d_in[1];
  // d_in[2] = out_lengths (compile-time constant 64)
  const float* W_ih  = (const float*)d_in[3];
  const float* b_ih  = (const float*)d_in[4];
  const float* W_hh  = (const float*)d_in[5];
  const float* b_hh  = (const float*)d_in[6];
  const float* W_out = (const float*)d_in[7];
  const float* b_out = (const float*)d_in[8];
  float*       out   = (float*)d_out;

  float*    hbuf0 = (float*)d_ws;                 // 128*512 f32
  float*    hbuf1 = hbuf0 + Bv * Hv;              // 128*512 f32
  unsigned* sync  = (unsigned*)(hbuf1 + Bv * Hv); // {cnt, gen}

  rnn_sync_init<<<1, 1, 0, stream>>>(sync);

  const size_t lds_bytes = (32 * Hv + 16 * Hv) * sizeof(float);  // 96 KB
  rnn_persistent<<<NWG, 256, lds_bytes, stream>>>(
      x, lengths, W_ih, b_ih, W_hh, b_hh, W_out, b_out,
      out, hbuf0, hbuf1, sync);
}